// erf_loss_18253611008036
// MI455X (gfx1250) — compile-verified
//
#include <hip/hip_runtime.h>
#include <hip/hip_bf16.h>
#include <stdint.h>

#define TPB  256   // 8 wave32s
#define CT   8     // classes per block tile
#define DMAX 512   // D for this problem (compile-time LDS sizing)

// erfc(x) for x >= 0. Numerical Recipes rational*exp approximation,
// |relative error| < 1.2e-7 — plenty for fp32 accumulation.
// Cost: 1 v_rcp_f32 + 11 v_fma_f32 + 1 fast exp (v_exp_f32) + 1 v_mul.
__device__ __forceinline__ float erfc_pos(float x) {
    float t = __builtin_amdgcn_rcpf(fmaf(0.5f, x, 1.0f));
    float p =              0.17087277f;
    p = fmaf(p, t, -0.82215223f);
    p = fmaf(p, t,  1.48851587f);
    p = fmaf(p, t, -1.13520398f);
    p = fmaf(p, t,  0.27886807f);
    p = fmaf(p, t, -0.18628806f);
    p = fmaf(p, t,  0.09678418f);
    p = fmaf(p, t,  0.37409196f);
    p = fmaf(p, t,  1.00002368f);
    p = fmaf(p, t, -1.26551223f);
    float z = fmaf(-x, x, p);         // -x^2 + poly(t)
    return t * __expf(z);             // underflows cleanly to 0 for large x
}

// CDNA5 async global->LDS DMA (VGLOBAL encoding, ASYNCcnt-tracked).
// VDST VGPR = LDS byte offset; VADDR = 64-bit global address.
// Low 32 bits of a generic pointer to __shared__ are the LDS byte offset
// (flat aperture check uses ADDR[63:32] only).
__device__ __forceinline__ void async_ld_b128(const float* g, uint32_t lds_off) {
    asm volatile("global_load_async_to_lds_b128 %0, %1, off"
                 :: "v"(lds_off), "v"((unsigned long long)(uintptr_t)g)
                 : "memory");
}

__device__ __forceinline__ void wait_async0() {
    asm volatile("s_wait_asynccnt 0" ::: "memory");
}

__global__ __launch_bounds__(TPB) void erf_loss_partial(
    const float* __restrict__ mu,      // [B, D]
    const float* __restrict__ stdv,    // [B, D]
    const float* __restrict__ weight,  // [C, D]
    const int*   __restrict__ label,   // [B]
    float* __restrict__ ws,            // [B * nTiles] partials
    int B, int C, int D)
{
    __shared__ float sW [CT * DMAX];   // 16 KB weight tile
    __shared__ float sMu[DMAX];        //  2 KB mu row
    __shared__ float sRs[DMAX];        //  2 KB 1/(sqrt2*(std+eps))
    __shared__ float sRed[TPB];

    const int tid = threadIdx.x;
    const int b   = blockIdx.y;
    const int c0  = blockIdx.x * CT;
    int ctn = C - c0; if (ctn > CT) ctn = CT;

    const int d4     = D >> 2;              // float4 chunks per row (=128)
    const int dshift = __builtin_ctz(d4);   // D is a power of two here
    const int total4 = ctn * d4;

    // ---- Stage operands via async global->LDS DMA (no VGPR round-trip) ----
    {
        const float* wsrc = weight + (size_t)c0 * D;
        for (int i = tid; i < total4; i += TPB)
            async_ld_b128(wsrc + (i << 2), (uint32_t)(uintptr_t)&sW[i << 2]);
        const float* msrc = mu + (size_t)b * D;
        for (int i = tid; i < d4; i += TPB)
            async_ld_b128(msrc + (i << 2), (uint32_t)(uintptr_t)&sMu[i << 2]);
    }
    // Reciprocal scale on the normal path (needs VALU anyway); overlaps DMA.
    for (int i = tid; i < D; i += TPB) {
        float s = stdv[(size_t)b * D + i] + 1e-8f;
        sRs[i] = 0.70710678118654752f * __builtin_amdgcn_rcpf(s);
    }
    wait_async0();
    __syncthreads();

    const int lab = label[b];
    const float4* W4 = reinterpret_cast<const float4*>(sW);
    const float4* M4 = reinterpret_cast<const float4*>(sMu);
    const float4* R4 = reinterpret_cast<const float4*>(sRs);

    float acc = 0.0f;
    if (d4 <= TPB) {
        // TPB is a multiple of d4: each thread owns ONE d-chunk j across all
        // rows -> hoist the mu/rs LDS loads out of the hot loop entirely.
        const int j     = tid & (d4 - 1);   // fixed chunk within a row
        const int cstep = TPB >> dshift;    // rows covered per sweep (=2)
        const int cbeg  = tid >> dshift;    // starting row (0 or 1)
        const float4 m = M4[j];
        const float4 r = R4[j];
        for (int c = cbeg; c < ctn; c += cstep) {
            float4 w = W4[(c << dshift) + j];   // 1 ds_load_b128 / iter
            float v0 = erfc_pos(fabsf(w.x - m.x) * r.x);
            float v1 = erfc_pos(fabsf(w.y - m.y) * r.y);
            float v2 = erfc_pos(fabsf(w.z - m.z) * r.z);
            float v3 = erfc_pos(fabsf(w.w - m.w) * r.w);
            float s4 = (v0 + v1) + (v2 + v3);
            // c == label: per element 1 + erf(y) = 2 - erfc(y) -> 8 - s4 / float4
            // c != label: per element erfc(y)                  -> s4
            acc += ((c0 + c) == lab) ? (8.0f - s4) : s4;
        }
    } else {
        // Generic fallback (not taken for D=512): flat index over the tile.
        for (int i = tid; i < total4; i += TPB) {
            const int c = i >> dshift;
            const int j = i & (d4 - 1);
            float4 w = W4[i];
            float4 m = M4[j];
            float4 r = R4[j];
            float v0 = erfc_pos(fabsf(w.x - m.x) * r.x);
            float v1 = erfc_pos(fabsf(w.y - m.y) * r.y);
            float v2 = erfc_pos(fabsf(w.z - m.z) * r.z);
            float v3 = erfc_pos(fabsf(w.w - m.w) * r.w);
            float s4 = (v0 + v1) + (v2 + v3);
            acc += ((c0 + c) == lab) ? (8.0f - s4) : s4;
        }
    }

    // Deterministic block reduction
    sRed[tid] = acc;
    __syncthreads();
    for (int s = TPB >> 1; s > 0; s >>= 1) {
        if (tid < s) sRed[tid] += sRed[tid + s];
        __syncthreads();
    }
    if (tid == 0) ws[(size_t)b * gridDim.x + blockIdx.x] = sRed[0];
}

__global__ __launch_bounds__(TPB) void erf_loss_final(
    const float* __restrict__ ws, int n, float* __restrict__ out, double scale)
{
    __shared__ double red[TPB];
    double a = 0.0;
    for (int i = threadIdx.x; i < n; i += TPB) a += (double)ws[i];
    red[threadIdx.x] = a;
    __syncthreads();
    for (int s = TPB >> 1; s > 0; s >>= 1) {
        if (threadIdx.x < s) red[threadIdx.x] += red[threadIdx.x + s];
        __syncthreads();
    }
    if (threadIdx.x == 0) out[0] = (float)(red[0] * scale);
}

extern "C" void kernel_launch(void* const* d_in, const int* in_sizes, int n_in,
                              void* d_out, int out_size, void* d_ws, size_t ws_size,
                              hipStream_t stream) {
    const float* mu     = (const float*)d_in[0];
    const float* stdv   = (const float*)d_in[1];
    const float* weight = (const float*)d_in[2];
    const int*   label  = (const int*)d_in[3];

    const int B = in_sizes[3];           // 128
    const int D = in_sizes[0] / B;       // 512
    const int C = in_sizes[2] / D;       // 1000
    const int nTiles = (C + CT - 1) / CT;

    float* ws = (float*)d_ws;            // nTiles*B floats (64 KB) of scratch

    dim3 grid(nTiles, B);
    erf_loss_partial<<<grid, TPB, 0, stream>>>(mu, stdv, weight, label, ws, B, C, D);

    const double scale = 1.0 / ((double)B * (double)C);
    erf_loss_final<<<1, TPB, 0, stream>>>(ws, nTiles * B, (float*)d_out, scale);
}